// UnweightedHadamardMixing_66176856096919
// MI455X (gfx1250) — compile-verified
//
#include <hip/hip_runtime.h>

// out[b,o,p] = sum_i W[o,i] * x[b,i,p],  W = H64/8, H64 = H4 (x) H16 (Sylvester).
// Stage 1 (WMMA): t[g] = (H16/8) @ x[16g:16g+16, tile]   -- 4 groups, K=16 each
//                 via 4x V_WMMA_F32_16X16X4_F32 per group (16 wmma/tile).
// Stage 2 (VALU): H4 butterfly across groups: 64 add/sub per tile.
// A-fragments for H16/8 are lane-invariant across tiles: 8 VGPRs, hoisted.

typedef __attribute__((ext_vector_type(2))) float v2f;
typedef __attribute__((ext_vector_type(8))) float v8f;

#define HWPIX 65536            // 256*256 pixels per (b, channel) plane
#define NCH   64
#define TILES_PER_BATCH 4096   // 65536 / 16

__global__ __launch_bounds__(256) void had_mix_wmma_f32(
    const float* __restrict__ x, float* __restrict__ out, int n_tiles) {
  const int tid  = threadIdx.x;
  const int lane = tid & 31;
  const int ln   = lane & 15;   // pixel within tile / A row / D col
  const int hi   = lane >> 4;   // lane-half: K pair select / D row half
  const int wave = tid >> 5;

  // A-fragments of H16/8, k-step kk in 0..3: lane holds A(ln, k) = H16[ln, i]/8
  // with i = 4*kk + 2*hi + {0,1}; H16[r,i] = (-1)^popc(r&i). Loop-invariant.
  v2f a[4];
#pragma unroll
  for (int kk = 0; kk < 4; ++kk) {
    const int i0 = kk * 4 + hi * 2;                       // even column
    const unsigned s0 = __builtin_popcount((unsigned)(ln & i0)) & 1u;
    const float a0 = s0 ? -0.125f : 0.125f;
    a[kk].x = a0;
    a[kk].y = (ln & 1) ? -a0 : a0;  // odd column flips by parity(ln&1)
  }

  const int waves_per_grid = (gridDim.x * blockDim.x) >> 5;
  int tile = blockIdx.x * (blockDim.x >> 5) + wave;

  for (; tile < n_tiles; tile += waves_per_grid) {
    const int b = tile >> 12;             // tile / TILES_PER_BATCH
    const int p = (tile & 4095) << 4;     // pixel base within batch plane

    const float* xb = x   + (size_t)b * NCH * HWPIX + (size_t)(p + ln);
    float*       ob = out + (size_t)b * NCH * HWPIX + (size_t)(p + ln);

    v8f t[4];
#pragma unroll
    for (int g = 0; g < 4; ++g) t[g] = (v8f){0.f,0.f,0.f,0.f,0.f,0.f,0.f,0.f};

    // Stage 1: t[g] = (H16/8) @ x[16g.., pix]  (16x16x16 via 4 wmma k-steps)
#pragma unroll
    for (int g = 0; g < 4; ++g) {
#pragma unroll
      for (int kk = 0; kk < 4; ++kk) {
        const int c0 = g * 16 + kk * 4 + hi * 2;   // even input channel
        v2f bfrag;
        bfrag.x = xb[(size_t)c0 * HWPIX];
        bfrag.y = xb[(size_t)c0 * HWPIX + HWPIX];
        t[g] = __builtin_amdgcn_wmma_f32_16x16x4_f32(
            /*neg_a=*/false, a[kk], /*neg_b=*/false, bfrag,
            /*c_mod=*/(short)0, t[g], /*reuse_a=*/false, /*reuse_b=*/false);
      }
    }

    // Stage 2: H4 butterfly across groups. H4[o_hi,g] = (-1)^popc(o_hi&g):
    //   o0 = t0+t1+t2+t3, o1 = t0-t1+t2-t3, o2 = t0+t1-t2-t3, o3 = t0-t1-t2+t3
    v8f u0 = t[0] + t[1];
    v8f u1 = t[0] - t[1];
    v8f u2 = t[2] + t[3];
    v8f u3 = t[2] - t[3];
    v8f og[4];
    og[0] = u0 + u2;
    og[1] = u1 + u3;
    og[2] = u0 - u2;
    og[3] = u1 - u3;

    // D layout: VGPR r, lanes 0-15 -> row r, lanes 16-31 -> row r+8
#pragma unroll
    for (int g = 0; g < 4; ++g) {
#pragma unroll
      for (int r = 0; r < 8; ++r) {
        ob[(size_t)(g * 16 + r + 8 * hi) * HWPIX] = og[g][r];
      }
    }
  }
}

extern "C" void kernel_launch(void* const* d_in, const int* in_sizes, int n_in,
                              void* d_out, int out_size, void* d_ws, size_t ws_size,
                              hipStream_t stream) {
  const float* x = (const float*)d_in[0];
  // d_in[1] (h_kernel) is the deterministic Hadamard/8; synthesized analytically.
  float* out = (float*)d_out;

  const int n_tiles = 32 * TILES_PER_BATCH;   // 131072 16-pixel tiles
  // 2048 blocks * 8 waves = 16384 waves; 131072 / 16384 = 8 tiles/wave exactly,
  // so the grid-stride loop is wave-uniform and EXEC stays all-ones for WMMA.
  had_mix_wmma_f32<<<dim3(2048), dim3(256), 0, stream>>>(x, out, n_tiles);
}